// Base_84980222919454
// MI455X (gfx1250) — compile-verified
//
#include <hip/hip_runtime.h>

typedef __attribute__((ext_vector_type(2))) float v2f;
typedef __attribute__((ext_vector_type(8))) float v8f;
typedef __attribute__((ext_vector_type(4))) unsigned int v4u;
typedef __attribute__((ext_vector_type(4))) int v4i;
typedef __attribute__((ext_vector_type(8))) int v8i;

// LDS accessed via raw workgroup-relative byte offsets in address_space(3), so the
// compiler cannot prove anything about the data the TDM engine deposited there
// (round 2 showed that a named __shared__ array gets its reads folded to undef,
// because tensor_load_to_lds only carries the LDS address as descriptor bits).
typedef __attribute__((address_space(3))) float lds_f;
typedef __attribute__((address_space(3))) v2f   lds_v2f;

__device__ __forceinline__ v2f lds_read_v2f(unsigned byte_off) {
    return *(const lds_v2f*)byte_off;      // ds_load_b64
}
__device__ __forceinline__ float lds_read_f(unsigned byte_off) {
    return *(const lds_f*)byte_off;        // ds_load_b32
}

#define BM 128
#define BN 64
#define KB 32
#define LDA 34                      /* A LDS row stride in floats: 32 + 2 pad (TDM pad) */
#define AS0 0
#define AS1 (BM * LDA)              /* float offsets inside the dynamic LDS block */
#define BS0 (2 * BM * LDA)
#define BS1 (2 * BM * LDA + KB * BN)
#define SMEM_FLOATS (2 * BM * LDA + 2 * KB * BN)   /* 12800 floats = 51200 B */
#define SMEM_BYTES  (SMEM_FLOATS * 4)

// Issue one 2-D TDM tile load: global (row-major, stride_elems f32 per row) -> LDS.
// Tensor dims are the *remaining* extents from gptr so OOB rows/cols zero-fill LDS.
// D# layout per CDNA5 ISA 8.3/8.4 (group0 128b, group1 256b; groups 2/3 zero => 2-D).
__device__ __forceinline__ void tdm_load_tile_2d(unsigned lds_byte_off, const float* gptr,
                                                 int tile_d0, int tile_d1,
                                                 int ext_d0, int ext_d1,
                                                 int stride_elems, unsigned pad_fields)
{
    unsigned long long ga = (unsigned long long)(size_t)gptr;

    v4u g0;
    g0.x = 1u;                                              // count=1, user descriptor
    g0.y = lds_byte_off;                                    // lds_addr
    g0.z = (unsigned)(ga & 0xFFFFFFFFu);                    // global_addr[31:0]
    g0.w = (unsigned)((ga >> 32) & 0x01FFFFFFull) | (2u << 30);  // global_addr[56:32] | type=2

    v8i g1;
    g1[0] = (int)((2u << 16) | pad_fields);                 // data_size=4B + pad cfg
    g1[1] = (int)(((unsigned)ext_d0 & 0xFFFFu) << 16);      // tensor_dim0[15:0] (atomic addr=0)
    g1[2] = (int)((((unsigned)ext_d0 >> 16) & 0xFFFFu) | (((unsigned)ext_d1 & 0xFFFFu) << 16));
    g1[3] = (int)((((unsigned)ext_d1 >> 16) & 0xFFFFu) | (((unsigned)tile_d0 & 0xFFFFu) << 16));
    g1[4] = (int)((unsigned)tile_d1 & 0xFFFFu);             // tile_dim1 (tile_dim2=0)
    g1[5] = stride_elems;                                   // tensor_dim0_stride[31:0]
    g1[6] = 0;                                              // stride hi / dim1_stride (2-D: unused)
    g1[7] = 0;

    v4i g2 = {0, 0, 0, 0};
    v4i g3 = {0, 0, 0, 0};
#if defined(__clang_major__) && (__clang_major__ >= 23)
    v8i gz = {0, 0, 0, 0, 0, 0, 0, 0};
    __builtin_amdgcn_tensor_load_to_lds(g0, g1, g2, g3, gz, 0);
#else
    __builtin_amdgcn_tensor_load_to_lds(g0, g1, g2, g3, 0);
#endif
}

// C[M,N] = A[M,K] @ B[K,N] (+ optional bias & relu), fp32 via V_WMMA_F32_16X16X4_F32.
// Block: 256 threads = 8 waves; block tile 128x64; wave tile 32x32 (2x2 WMMA subtiles).
// K-panels (A 128x32 padded to stride 34, B 32x64) staged in LDS by the Tensor Data
// Mover, double buffered; wave 0 issues next panel's DMA while all waves compute.
__global__ __launch_bounds__(256)
void gemm_wmma_tdm_kernel(const float* __restrict__ A, const float* __restrict__ B,
                          const float* __restrict__ bias, float* __restrict__ C,
                          int M, int N, int K, int fuse_bias_relu)
{
    const int lane = threadIdx.x & 31;
    const int wave = threadIdx.x >> 5;
    const int half = lane >> 4;      // 0: lanes 0-15, 1: lanes 16-31
    const int lq   = lane & 15;
    const int wm   = wave & 3;       // 4 row tiles of 32
    const int wn   = wave >> 2;      // 2 col tiles of 32

    const int row0 = blockIdx.y * BM;
    const int col0 = blockIdx.x * BN;

    // pad: interval 32 DWORDs (code 4), amount 2 DWORDs (code 1) -> LDS stride 34 floats
    const unsigned padA = (1u << 20) | (4u << 22) | (1u << 25);

    v8f acc[4];
    v8f vzero = {};
    acc[0] = vzero; acc[1] = vzero; acc[2] = vzero; acc[3] = vzero;

    const int numP = (K + KB - 1) / KB;

    if (wave == 0) {
        tdm_load_tile_2d(AS0 * 4u, A + (size_t)row0 * K, KB, BM, K, M - row0, K, padA);
        tdm_load_tile_2d(BS0 * 4u, B + col0,             BN, KB, N - col0, K, N, 0u);
    }

    // Per-lane fragment byte offsets within a panel buffer
    const unsigned aoff0 = (unsigned)((wm * 32 + lq) * LDA) * 4u;        // A frag row
    const unsigned aoff1 = (unsigned)((wm * 32 + 16 + lq) * LDA) * 4u;
    const unsigned boff0 = (unsigned)(wn * 32 + lq) * 4u;                // B frag col
    const unsigned boff1 = (unsigned)(wn * 32 + 16 + lq) * 4u;

    for (int p = 0; p < numP; ++p) {
        __builtin_amdgcn_s_wait_tensorcnt(0);   // no-op for waves with TENSORcnt==0
        __syncthreads();                         // panel p visible; panel p-1 buffer free

        if (wave == 0 && (p + 1) < numP) {
            const int k1 = (p + 1) * KB;
            const unsigned a1 = ((p + 1) & 1) ? AS1 : AS0;
            const unsigned b1 = ((p + 1) & 1) ? BS1 : BS0;
            tdm_load_tile_2d(a1 * 4u, A + (size_t)row0 * K + k1, KB, BM, K - k1, M - row0, K, padA);
            tdm_load_tile_2d(b1 * 4u, B + (size_t)k1 * N + col0, BN, KB, N - col0, K - k1, N, 0u);
        }

        const unsigned AsB = (unsigned)(((p & 1) ? AS1 : AS0) * 4);
        const unsigned BsB = (unsigned)(((p & 1) ? BS1 : BS0) * 4);

        #pragma unroll
        for (int ks = 0; ks < KB / 4; ++ks) {
            const int kk = ks * 4 + 2 * half;    // lane holds K = 2*half + {0,1}
            v2f a0 = lds_read_v2f(AsB + aoff0 + (unsigned)kk * 4u);   // stride-34: conflict-free
            v2f a1 = lds_read_v2f(AsB + aoff1 + (unsigned)kk * 4u);
            v2f b0, b1;
            b0.x = lds_read_f(BsB + (unsigned)(kk * BN) * 4u + boff0);
            b0.y = lds_read_f(BsB + (unsigned)((kk + 1) * BN) * 4u + boff0);
            b1.x = lds_read_f(BsB + (unsigned)(kk * BN) * 4u + boff1);
            b1.y = lds_read_f(BsB + (unsigned)((kk + 1) * BN) * 4u + boff1);

            acc[0] = __builtin_amdgcn_wmma_f32_16x16x4_f32(false, a0, false, b0, (short)0, acc[0], false, false);
            acc[1] = __builtin_amdgcn_wmma_f32_16x16x4_f32(false, a0, false, b1, (short)0, acc[1], false, false);
            acc[2] = __builtin_amdgcn_wmma_f32_16x16x4_f32(false, a1, false, b0, (short)0, acc[2], false, false);
            acc[3] = __builtin_amdgcn_wmma_f32_16x16x4_f32(false, a1, false, b1, (short)0, acc[3], false, false);
        }
    }

    // C/D layout: VGPR v, lane -> M = 8*half + v, N = lane%16
    #pragma unroll
    for (int t = 0; t < 4; ++t) {
        const int rt = t >> 1, ct = t & 1;
        const int n = col0 + wn * 32 + ct * 16 + lq;
        if (n >= N) continue;
        const float bval = fuse_bias_relu ? bias[n] : 0.0f;
        #pragma unroll
        for (int v = 0; v < 8; ++v) {
            const int mr = row0 + wm * 32 + rt * 16 + half * 8 + v;
            if (mr >= M) continue;
            float x = acc[t][v];
            if (fuse_bias_relu) x = fmaxf(x + bval, 0.0f);
            C[(size_t)mr * (size_t)N + n] = x;
        }
    }
}

// Z[dst[e], :] += Y[src[e], :]   -- one block per edge, float4 reads, f32 atomics
__global__ __launch_bounds__(256)
void scatter_add_kernel(const float* __restrict__ Y, const int* __restrict__ src,
                        const int* __restrict__ dst, float* __restrict__ Z, int D)
{
    const int e = blockIdx.x;
    const int c = threadIdx.x << 2;
    if (c >= D) return;
    const int s = src[e];
    const int d = dst[e];
    const float4 v = *(const float4*)(Y + (size_t)s * D + c);
    float* zp = Z + (size_t)d * D + c;
    atomicAdd(zp + 0, v.x);
    atomicAdd(zp + 1, v.y);
    atomicAdd(zp + 2, v.z);
    atomicAdd(zp + 3, v.w);
}

// Z = relu(Z + b) row-wise  -- one block per node
__global__ __launch_bounds__(256)
void bias_relu_kernel(float* __restrict__ Z, const float* __restrict__ bias, int D)
{
    const int row = blockIdx.x;
    const int c = threadIdx.x << 2;
    if (c >= D) return;
    float4 v = *(float4*)(Z + (size_t)row * D + c);
    const float4 b = *(const float4*)(bias + c);
    v.x = fmaxf(v.x + b.x, 0.0f);
    v.y = fmaxf(v.y + b.y, 0.0f);
    v.z = fmaxf(v.z + b.z, 0.0f);
    v.w = fmaxf(v.w + b.w, 0.0f);
    *(float4*)(Z + (size_t)row * D + c) = v;
}

extern "C" void kernel_launch(void* const* d_in, const int* in_sizes, int n_in,
                              void* d_out, int out_size, void* d_ws, size_t ws_size,
                              hipStream_t stream)
{
    const float* features = (const float*)d_in[0];
    const int*   src      = (const int*)  d_in[1];
    const int*   dst      = (const int*)  d_in[2];
    const float* W1       = (const float*)d_in[3];
    const float* b1       = (const float*)d_in[4];
    const float* W2       = (const float*)d_in[5];
    const float* b2       = (const float*)d_in[6];
    const float* W3       = (const float*)d_in[7];
    const float* b3       = (const float*)d_in[8];
    float* out = (float*)d_out;

    const int NN = 20000, E = 200000;
    const int D_IN = 1433, D_H1 = 1000, D_H2 = 500, D_OUT = 7;

    float* buf0 = (float*)d_ws;                 // 80 MB: Y1, then Y2
    float* buf1 = buf0 + (size_t)NN * D_H1;     // 80 MB: Z1/H1, then Z2/H2

    const dim3 blk(256);

    // Layer 1: Y1 = X @ W1  (agg-GEMM commutation: agg(X)@W1 == agg(X@W1))
    {
        dim3 grid((D_H1 + BN - 1) / BN, (NN + BM - 1) / BM);
        gemm_wmma_tdm_kernel<<<grid, blk, SMEM_BYTES, stream>>>(features, W1, nullptr, buf0,
                                                                NN, D_H1, D_IN, 0);
    }
    hipMemsetAsync(buf1, 0, (size_t)NN * D_H1 * sizeof(float), stream);
    scatter_add_kernel<<<E, 256, 0, stream>>>(buf0, src, dst, buf1, D_H1);
    bias_relu_kernel<<<NN, 256, 0, stream>>>(buf1, b1, D_H1);

    // Layer 2: Y2 = H1 @ W2
    {
        dim3 grid((D_H2 + BN - 1) / BN, (NN + BM - 1) / BM);
        gemm_wmma_tdm_kernel<<<grid, blk, SMEM_BYTES, stream>>>(buf1, W2, nullptr, buf0,
                                                                NN, D_H2, D_H1, 0);
    }
    hipMemsetAsync(buf1, 0, (size_t)NN * D_H2 * sizeof(float), stream);
    scatter_add_kernel<<<E, 128, 0, stream>>>(buf0, src, dst, buf1, D_H2);
    bias_relu_kernel<<<NN, 128, 0, stream>>>(buf1, b2, D_H2);

    // Layer 3: out = relu(H2 @ W3 + b3)  (N=7 zero-padded by TDM into one 64-wide tile)
    {
        dim3 grid(1, (NN + BM - 1) / BM);
        gemm_wmma_tdm_kernel<<<grid, blk, SMEM_BYTES, stream>>>(buf1, W3, b3, out,
                                                                NN, D_OUT, D_H2, 1);
    }
}